// MoE_17738214933199
// MI455X (gfx1250) — compile-verified
//
#include <hip/hip_runtime.h>

typedef __attribute__((ext_vector_type(2))) float v2f;
typedef __attribute__((ext_vector_type(8))) float v8f;

#define T_TOK   16384
#define DDIM    1024
#define EEXP    8
#define HDIM    8
#define NCOLS   144      // 8 gate + 64 g + 64 p + 8 pad -> 9 tiles of 16
#define NTILES  9
#define KCHUNK  64       // K rows staged per chunk (32 K-pairs)
#define KPAIRS  (KCHUNK/2)
#define OWP_STR 1040     // padded ow pair-stride: 2*1040 % 64 == 32 -> disjoint banks

__device__ __forceinline__ v8f wmma_f32_16x16x4(v2f a, v2f b, v8f c) {
    // D = A(16x4,f32) * B(4x16,f32) + C(16x16,f32)
    return __builtin_amdgcn_wmma_f32_16x16x4_f32(
        /*neg_a=*/false, a, /*neg_b=*/false, b,
        /*c_mod=*/(short)0, c, /*reuse_a=*/false, /*reuse_b=*/false);
}

// ---------------------------------------------------------------------------
// Kernel 1: fused router + dense expert projections + gating.
// Each wave (wave32) owns a 16-token tile. Per block: 8 waves = 128 tokens.
// B staged in LDS as K-pairs (float2) -> one aligned ds_load_b64 per fragment.
// ---------------------------------------------------------------------------
__global__ __launch_bounds__(256) void moe_proj_kernel(
    const float* __restrict__ x,        // [T, D]
    const float* __restrict__ gate_w,   // [D, E]
    const float* __restrict__ gw,       // [E, D, H]
    const float* __restrict__ pw,       // [E, D, H]
    float* __restrict__ hw,             // [T, 64] workspace (combine-weighted h)
    float* __restrict__ logits_out,     // [T, E]
    float* __restrict__ blacc)          // [16]: cnt[8], sumprob[8]
{
    __shared__ v2f   wc2[KPAIRS * NCOLS];       // 32*144*8B = 36 KB
    __shared__ float sc_log[8][16 * 16];        // per-wave logits scratch
    __shared__ float sc_cmb[8][16 * 8];         // per-wave combine weights
    __shared__ float bl_s[16];                  // block-level loss accumulators

    const int tid  = threadIdx.x;
    const int wave = tid >> 5;
    const int lane = tid & 31;
    const int tok0 = (blockIdx.x * 8 + wave) * 16;

    if (tid < 16) bl_s[tid] = 0.0f;

    const int mrow = lane & 15;
    const int phalf = lane >> 4;                // 0 or 1
    const int koff  = phalf * 2;                // K offset within 4-step
    const float* xrow = x + (size_t)(tok0 + mrow) * DDIM;

    v8f acc[NTILES];
#pragma unroll
    for (int t = 0; t < NTILES; ++t) acc[t] = (v8f){0,0,0,0,0,0,0,0};

    for (int kc = 0; kc < DDIM; kc += KCHUNK) {
        __syncthreads();                        // previous chunk fully consumed
        // Cooperative stage: wc2[kp][n] = { Wcat[kc+2kp][n], Wcat[kc+2kp+1][n] }
        for (int idx = tid; idx < KPAIRS * NCOLS; idx += 256) {
            const int kp = idx / NCOLS;
            const int n  = idx - kp * NCOLS;
            const int kg = kc + 2 * kp;
            v2f v;
            if (n < 8) {
                v.x = gate_w[kg * EEXP + n];
                v.y = gate_w[(kg + 1) * EEXP + n];
            } else if (n < 72) {
                const int c = n - 8;
                const float* b = gw + (size_t)(c >> 3) * (DDIM*HDIM) + (c & 7);
                v.x = b[kg * HDIM]; v.y = b[(kg + 1) * HDIM];
            } else if (n < 136) {
                const int c = n - 72;
                const float* b = pw + (size_t)(c >> 3) * (DDIM*HDIM) + (c & 7);
                v.x = b[kg * HDIM]; v.y = b[(kg + 1) * HDIM];
            } else {
                v.x = 0.0f; v.y = 0.0f;
            }
            wc2[idx] = v;
        }
        __syncthreads();

        // A fragments for this K-chunk: 16 steps x v2f (b64 global loads).
        v2f afrag[16];
#pragma unroll
        for (int s = 0; s < 16; ++s)
            afrag[s] = *(const v2f*)(xrow + kc + 4 * s + koff);

#pragma unroll          // FULL unroll: constant afrag indices, no v_movrels
        for (int s = 0; s < 16; ++s) {
            const int kp = 2 * s + phalf;       // this half-wave's K-pair
#pragma unroll
            for (int t = 0; t < NTILES; ++t) {
                const int n = t * 16 + mrow;
                const v2f b = wc2[kp * NCOLS + n];   // single ds_load_b64
                acc[t] = wmma_f32_16x16x4(afrag[s], b, acc[t]);
            }
        }
    }

    // --- router: logits live in acc[0] cols 0..7 ---
    float* sc = &sc_log[wave][0];
    const int mhi = phalf * 8;
#pragma unroll
    for (int r = 0; r < 8; ++r)
        sc[(r + mhi) * 16 + mrow] = acc[0][r];
    asm volatile("s_wait_dscnt 0x0" ::: "memory");

    float* cb = &sc_cmb[wave][0];
    if (lane < 16) {
        float lg[8], pr[8];
        float mx = -3.4e38f;
#pragma unroll
        for (int e = 0; e < EEXP; ++e) { lg[e] = sc[lane * 16 + e]; mx = fmaxf(mx, lg[e]); }
        float sum = 0.0f;
#pragma unroll
        for (int e = 0; e < EEXP; ++e) { pr[e] = __expf(lg[e] - mx); sum += pr[e]; }
        const float inv = 1.0f / sum;
#pragma unroll
        for (int e = 0; e < EEXP; ++e) pr[e] *= inv;

        // top-2, earliest index wins ties (matches jax.lax.top_k)
        int i1 = 0;
#pragma unroll
        for (int e = 1; e < EEXP; ++e) if (pr[e] > pr[i1]) i1 = e;
        int i2 = (i1 == 0) ? 1 : 0;
#pragma unroll
        for (int e = 0; e < EEXP; ++e) if (e != i1 && pr[e] > pr[i2]) i2 = e;
        const float wn = 1.0f / (pr[i1] + pr[i2]);

#pragma unroll
        for (int e = 0; e < EEXP; ++e) cb[lane * 8 + e] = 0.0f;
        cb[lane * 8 + i1] = pr[i1] * wn;
        cb[lane * 8 + i2] = pr[i2] * wn;

        // router_logits out (two 16B stores per token row)
        float* lo = logits_out + (size_t)(tok0 + lane) * EEXP;
        *(float4*)(lo)     = make_float4(lg[0], lg[1], lg[2], lg[3]);
        *(float4*)(lo + 4) = make_float4(lg[4], lg[5], lg[6], lg[7]);

        // load-balance stats into block-level LDS accumulators
        atomicAdd(&bl_s[i1], 1.0f);
        atomicAdd(&bl_s[i2], 1.0f);
#pragma unroll
        for (int e = 0; e < EEXP; ++e) atomicAdd(&bl_s[8 + e], pr[e]);
    }
    asm volatile("s_wait_dscnt 0x0" ::: "memory");

    // --- hw[t, e*8+j] = g * silu(p) * combine[t,e] ---
    // g cols n=8..71 live in tiles 0..4; matching p col is n+64 -> tile+4, same lane.
#pragma unroll
    for (int t = 0; t < 5; ++t) {
        const int n = t * 16 + mrow;
        const int c = n - 8;
        if (c >= 0 && c < 64) {
            const int e = c >> 3;
#pragma unroll
            for (int r = 0; r < 8; ++r) {
                const int m = r + mhi;
                const float g = acc[t][r];
                const float p = acc[t + 4][r];
                const float sp = p * __frcp_rn(1.0f + __expf(-p));  // silu
                hw[(size_t)(tok0 + m) * 64 + c] = g * sp * cb[m * 8 + e];
            }
        }
    }

    __syncthreads();
    if (tid < 16) atomicAdd(&blacc[tid], bl_s[tid]);
}

// ---------------------------------------------------------------------------
// Kernel 2: final[T,1024] = HW[T,64] @ ow[(E*H)=64, D=1024]
// ow staged once per block into dynamic LDS as K-pairs (float2), padded stride.
// ---------------------------------------------------------------------------
__global__ __launch_bounds__(256) void moe_out_kernel(
    const float* __restrict__ hw,    // [T, 64]
    const float* __restrict__ ow,    // [64, 1024] (== [E,H,D] flattened)
    float* __restrict__ out)         // [T, 1024]
{
    extern __shared__ v2f ows2[];    // [32 pairs][OWP_STR] (~260 KB)

    const int tid  = threadIdx.x;
    const int wave = tid >> 5;
    const int lane = tid & 31;
    const int tok0 = (blockIdx.x * 8 + wave) * 16;

    // Cooperative stage: ows2[kp][d] = { ow[2kp][d], ow[2kp+1][d] }
    for (int idx = tid; idx < 32 * 1024; idx += 256) {
        const int kp = idx >> 10;
        const int d  = idx & 1023;
        v2f v;
        v.x = ow[(size_t)(2 * kp)     * 1024 + d];
        v.y = ow[(size_t)(2 * kp + 1) * 1024 + d];
        ows2[kp * OWP_STR + d] = v;
    }
    __syncthreads();

    const int m     = lane & 15;
    const int phalf = lane >> 4;
    const int koff  = phalf * 2;

    v2f a[16];
    const float* hr = hw + (size_t)(tok0 + m) * 64;
#pragma unroll
    for (int s = 0; s < 16; ++s)
        a[s] = *(const v2f*)(hr + 4 * s + koff);

    for (int nt = 0; nt < 64; ++nt) {
        v8f c = (v8f){0,0,0,0,0,0,0,0};
        const int n = nt * 16 + m;
#pragma unroll
        for (int s = 0; s < 16; ++s) {
            const int kp = 2 * s + phalf;
            const v2f b = ows2[kp * OWP_STR + n];    // single ds_load_b64
            c = wmma_f32_16x16x4(a[s], b, c);
        }
        const int mb = phalf * 8;
#pragma unroll
        for (int r = 0; r < 8; ++r)
            out[(size_t)(tok0 + r + mb) * DDIM + nt * 16 + m] = c[r];
    }
}

// ---------------------------------------------------------------------------
// Kernel 3: bl_loss = E * sum_e cnt[e]*sumprob[e] / T^2
// ---------------------------------------------------------------------------
__global__ void moe_loss_kernel(const float* __restrict__ blacc,
                                float* __restrict__ loss_out)
{
    if (threadIdx.x == 0 && blockIdx.x == 0) {
        float s = 0.0f;
#pragma unroll
        for (int e = 0; e < EEXP; ++e) s += blacc[e] * blacc[8 + e];
        *loss_out = s * (float)EEXP / ((float)T_TOK * (float)T_TOK);
    }
}

extern "C" void kernel_launch(void* const* d_in, const int* in_sizes, int n_in,
                              void* d_out, int out_size, void* d_ws, size_t ws_size,
                              hipStream_t stream) {
    const float* x      = (const float*)d_in[0];   // [4,4096,1024]
    const float* gate_w = (const float*)d_in[1];   // [1024,8]
    const float* gw     = (const float*)d_in[2];   // [8,1024,8]
    const float* pw     = (const float*)d_in[3];   // [8,1024,8]
    const float* ow     = (const float*)d_in[4];   // [8,8,1024]
    // d_in[5] = top_k (=2, hardcoded)

    float* out_final  = (float*)d_out;                       // 16,777,216 f32
    float* out_logits = out_final + (size_t)T_TOK * DDIM;    // 131,072 f32
    float* out_loss   = out_logits + (size_t)T_TOK * EEXP;   // 1 f32

    float* hw    = (float*)d_ws;                             // [T,64]
    float* blacc = hw + (size_t)T_TOK * 64;                  // 16 f32

    hipMemsetAsync(blacc, 0, 16 * sizeof(float), stream);

    const int blocks = T_TOK / 128;  // 8 waves x 16 tokens per block
    moe_proj_kernel<<<blocks, 256, 0, stream>>>(x, gate_w, gw, pw, hw, out_logits, blacc);
    moe_out_kernel<<<blocks, 256, 32 * OWP_STR * sizeof(v2f), stream>>>(hw, ow, out_final);
    moe_loss_kernel<<<1, 32, 0, stream>>>(blacc, out_loss);
}